// LightGCNModel_41137196761091
// MI455X (gfx1250) — compile-verified
//
#include <hip/hip_runtime.h>

// LightGCN on MI455X (gfx1250, wave32).
// Phases:
//  1) copy tables into ping buffer + sum buffer
//  2) 3x { memset agg; edge scatter-add (f32 atomics, L2-resident tables);
//          per-row L2-normalize (1 wave32/row) + accumulate into sums }
//  3) batched dot of 16384 (user,item) pairs via V_WMMA_F32_16X16X4_F32
//     (16 rows per wave; diagonal of 16x16 D gives the dots), scaled by 1/16.
//
// Workspace: 64*(3*NUM_USERS + 3*NUM_ITEMS)*4B = ~230.4 MB in d_ws.

#define NUM_USERS 200000
#define NUM_ITEMS 100000
#define DIM       64
#define NUM_EDGES 2000000
#define BATCH     16384
#define NLAYERS   3

typedef __attribute__((ext_vector_type(2))) float v2f;
typedef __attribute__((ext_vector_type(8))) float v8f;

// ---- init: table -> emb buffer and sum buffer (float4 streaming copy) ----
__global__ void copy2_f4(const float4* __restrict__ src, float4* __restrict__ a,
                         float4* __restrict__ b, int n4) {
  int i = blockIdx.x * blockDim.x + threadIdx.x;
  if (i < n4) {
    float4 v = src[i];
    a[i] = v;
    b[i] = v;
  }
}

// ---- edge scatter-add: both directions in one pass ----
// thread t: edge e = t/16, dim-chunk c = t%16 (4 floats per chunk).
// Gathers are float4 (one 256B row per edge endpoint, L2-resident);
// scatter is 4 f32 global atomics per direction.
__global__ void scatter_edges(const int* __restrict__ u_nodes,
                              const int* __restrict__ i_nodes,
                              const float* __restrict__ uemb,
                              const float* __restrict__ iemb,
                              float* __restrict__ uagg,
                              float* __restrict__ iagg) {
  unsigned t = blockIdx.x * blockDim.x + threadIdx.x;
  unsigned e = t >> 4;
  if (e >= NUM_EDGES) return;
  int c4 = (int)(t & 15u) * 4;

  int u  = u_nodes[e];
  int it = i_nodes[e];

  const float4 iv = *(const float4*)(iemb + (size_t)it * DIM + c4);
  const float4 uv = *(const float4*)(uemb + (size_t)u  * DIM + c4);

  float* up = uagg + (size_t)u  * DIM + c4;
  float* ip = iagg + (size_t)it * DIM + c4;

  atomicAdd(up + 0, iv.x); atomicAdd(up + 1, iv.y);
  atomicAdd(up + 2, iv.z); atomicAdd(up + 3, iv.w);
  atomicAdd(ip + 0, uv.x); atomicAdd(ip + 1, uv.y);
  atomicAdd(ip + 2, uv.z); atomicAdd(ip + 3, uv.w);
}

// ---- per-row L2 normalize (in place) + accumulate into sum ----
// One wave32 per row; lane handles 2 consecutive floats; shuffle-tree reduce.
__global__ void normalize_acc(float* __restrict__ uagg, float* __restrict__ usum,
                              float* __restrict__ iagg, float* __restrict__ isum) {
  unsigned gtid = blockIdx.x * blockDim.x + threadIdx.x;
  unsigned wave = gtid >> 5;
  int lane = (int)(threadIdx.x & 31u);

  float* agg;
  float* sum;
  size_t row;
  if (wave < NUM_USERS) {
    agg = uagg; sum = usum; row = wave;
  } else if (wave < NUM_USERS + NUM_ITEMS) {
    agg = iagg; sum = isum; row = wave - NUM_USERS;
  } else {
    return;
  }

  size_t off = row * DIM + (size_t)lane * 2;
  float2 v = *(const float2*)(agg + off);
  float ss = v.x * v.x + v.y * v.y;
#pragma unroll
  for (int m = 16; m > 0; m >>= 1) ss += __shfl_xor(ss, m, 32);
  float inv = 1.0f / fmaxf(sqrtf(ss), 1e-12f);
  v.x *= inv; v.y *= inv;
  *(float2*)(agg + off) = v;           // becomes next layer's input embedding

  float2 s = *(const float2*)(sum + off);
  s.x += v.x; s.y += v.y;
  *(float2*)(sum + off) = s;
}

// ---- final batched dot via V_WMMA_F32_16X16X4_F32 ----
// Wave handles 16 batch rows. A[m][k] = usum[uidx[m]][k], B[k][n] = isum[iidx[n]][k].
// D = A x B accumulated over K=64 in 16 chunks of 4; diagonal D[m][m] is the dot.
// A-frag layout (16x4 f32): lane L -> row L%16, K-pair base 2*(L/16) within chunk.
// B-frag (4x16 f32) has the mirrored layout, so both fragments gather identically.
__global__ void final_dot_wmma(const float* __restrict__ usum,
                               const float* __restrict__ isum,
                               const int* __restrict__ uidx,
                               const int* __restrict__ iidx,
                               float* __restrict__ out) {
  unsigned wave = (blockIdx.x * blockDim.x + threadIdx.x) >> 5;
  int lane = (int)(threadIdx.x & 31u);
  int b0 = (int)wave * 16;                 // 1024 waves cover BATCH=16384

  int row = lane & 15;
  int kh  = (lane >> 4) * 2;               // 0 for lanes 0-15, 2 for lanes 16-31

  int ui = uidx[b0 + row];
  int ii = iidx[b0 + row];
  const float* up = usum + (size_t)ui * DIM + kh;
  const float* ip = isum + (size_t)ii * DIM + kh;

  v8f c = {};
#pragma unroll
  for (int kc = 0; kc < 16; ++kc) {
    v2f a, b;
    a.x = up[kc * 4 + 0];
    a.y = up[kc * 4 + 1];
    b.x = ip[kc * 4 + 0];
    b.y = ip[kc * 4 + 1];
    // (neg_a, A, neg_b, B, c_mod, C, reuse_a, reuse_b)
    c = __builtin_amdgcn_wmma_f32_16x16x4_f32(false, a, false, b,
                                              (short)0, c, false, false);
  }

  // Diagonal extraction per C/D layout:
  //   m in 0..7  -> VGPR m,     lane m
  //   m in 8..15 -> VGPR m-8,   lane m+16
  int sel = (lane < 16) ? lane : (lane - 24);
  float diag = c[0];
#pragma unroll
  for (int r = 1; r < 8; ++r)
    if (sel == r) diag = c[r];

  if (lane < 8) {
    out[b0 + lane] = diag * 0.0625f;            // /(NLAYERS+1)^2 = /16
  } else if (lane >= 24) {
    out[b0 + (lane - 16)] = diag * 0.0625f;
  }
}

extern "C" void kernel_launch(void* const* d_in, const int* in_sizes, int n_in,
                              void* d_out, int out_size, void* d_ws, size_t ws_size,
                              hipStream_t stream) {
  const float* user_table   = (const float*)d_in[0];
  const float* item_table   = (const float*)d_in[1];
  const int*   user_indices = (const int*)d_in[2];
  const int*   item_indices = (const int*)d_in[3];
  const int*   edge_index   = (const int*)d_in[4];
  const int*   u_nodes = edge_index;              // edge_index[0,:]
  const int*   i_nodes = edge_index + NUM_EDGES;  // edge_index[1,:]
  float* out = (float*)d_out;

  const size_t UE = (size_t)NUM_USERS * DIM;
  const size_t IE = (size_t)NUM_ITEMS * DIM;
  float* ws   = (float*)d_ws;
  float* U0   = ws;            // user emb ping
  float* U1   = U0 + UE;       // user emb pong / agg
  float* USUM = U1 + UE;
  float* I0   = USUM + UE;     // item emb ping
  float* I1   = I0 + IE;       // item emb pong / agg
  float* ISUM = I1 + IE;

  // init: emb = table, sum = table
  {
    int nu4 = (int)(UE / 4), ni4 = (int)(IE / 4);
    copy2_f4<<<(nu4 + 255) / 256, 256, 0, stream>>>(
        (const float4*)user_table, (float4*)U0, (float4*)USUM, nu4);
    copy2_f4<<<(ni4 + 255) / 256, 256, 0, stream>>>(
        (const float4*)item_table, (float4*)I0, (float4*)ISUM, ni4);
  }

  float* uping[2] = {U0, U1};
  float* iping[2] = {I0, I1};
  for (int l = 0; l < NLAYERS; ++l) {
    float* ue = uping[l & 1];
    float* ua = uping[(l + 1) & 1];
    float* ie = iping[l & 1];
    float* ia = iping[(l + 1) & 1];

    hipMemsetAsync(ua, 0, UE * sizeof(float), stream);
    hipMemsetAsync(ia, 0, IE * sizeof(float), stream);

    unsigned nthreads = (unsigned)NUM_EDGES * 16u;           // 32M
    scatter_edges<<<(nthreads + 255u) / 256u, 256, 0, stream>>>(
        u_nodes, i_nodes, ue, ie, ua, ia);

    unsigned nwaves = NUM_USERS + NUM_ITEMS;                 // 300000 rows
    normalize_acc<<<(nwaves * 32u + 255u) / 256u, 256, 0, stream>>>(
        ua, USUM, ia, ISUM);
  }

  // final dot: 1024 waves (16 pairs each) -> 32768 threads
  final_dot_wmma<<<(BATCH / 16) * 32 / 256, 256, 0, stream>>>(
      USUM, ISUM, user_indices, item_indices, out);
}